// GradientbasedLoss_45775761440941
// MI455X (gfx1250) — compile-verified
//
#include <hip/hip_runtime.h>
#include <hip/hip_bf16.h>

typedef __attribute__((ext_vector_type(2))) float v2f;
typedef __attribute__((ext_vector_type(8))) float v8f;
typedef __attribute__((ext_vector_type(4))) unsigned int u32x4;
typedef __attribute__((ext_vector_type(8))) int i32x8;
typedef __attribute__((ext_vector_type(4))) int i32x4;

#define D_FEAT 128
#define EPW 4              // edges per wave (fallback kernel)
#define WAVES_PER_BLOCK 8  // fallback kernel
#define TDM_ROWS 8         // rows per tensor_load_to_lds gather (32-bit index mode max)
#define TDM_WAVES 4        // waves per block in TDM kernel (32KB LDS/block)

__device__ __forceinline__ float wave_reduce_sum(float v) {
#pragma unroll
  for (int off = 16; off > 0; off >>= 1) v += __shfl_xor(v, off, 32);
  return v;
}
__device__ __forceinline__ float wave_reduce_max(float v) {
#pragma unroll
  for (int off = 16; off > 0; off >>= 1) v = fmaxf(v, __shfl_xor(v, off, 32));
  return v;
}

// ---------------------------------------------------------------------------
__global__ void zero_ws_kernel(float* p, int n) {
  int i = blockIdx.x * blockDim.x + threadIdx.x;
  if (i < n) p[i] = 0.0f;
}

// ---------------------------------------------------------------------------
// diff = pred - data (stored if STORE) fused with SSE = sum(diff^2).
// WMMA identity: for V_WMMA_F32_16X16X4_F32, passing the same VGPR pair as
// A and B makes B == A^T, so D = A*A^T + C accumulates row sums-of-squares
// on the diagonal. Two independent accumulator chains for ILP; loop is
// wave-uniform so EXEC is all-1s at every WMMA.
// ---------------------------------------------------------------------------
template <bool STORE>
__global__ void diff_sse_kernel(const float* __restrict__ pred,
                                const float* __restrict__ data,
                                float* __restrict__ diff, float* __restrict__ sse,
                                int tiles) {
  const int lane = threadIdx.x & 31;
  const int wave = blockIdx.x * (blockDim.x >> 5) + (threadIdx.x >> 5);
  const int nwaves = gridDim.x * (blockDim.x >> 5);

  v8f c0 = {}, c1 = {};
  for (int t = wave * 2; t < tiles; t += nwaves * 2) {   // tiles is even
    size_t base0 = (size_t)t * 64 + (size_t)lane * 2;
    size_t base1 = base0 + 64;
    float2 p0 = *(const float2*)(pred + base0);
    float2 q0 = *(const float2*)(data + base0);
    float2 p1 = *(const float2*)(pred + base1);
    float2 q1 = *(const float2*)(data + base1);
    float2 d0, d1;
    d0.x = p0.x - q0.x; d0.y = p0.y - q0.y;
    d1.x = p1.x - q1.x; d1.y = p1.y - q1.y;
    if constexpr (STORE) {
      *(float2*)(diff + base0) = d0;
      *(float2*)(diff + base1) = d1;
    }
    v2f a0; a0.x = d0.x; a0.y = d0.y;
    v2f a1; a1.x = d1.x; a1.y = d1.y;
    c0 = __builtin_amdgcn_wmma_f32_16x16x4_f32(false, a0, false, a0, (short)0, c0, false, false);
    c1 = __builtin_amdgcn_wmma_f32_16x16x4_f32(false, a1, false, a1, (short)0, c1, false, false);
  }
  v8f c = c0 + c1;

  // Diagonal D[i][i]: i=0..7 -> lane i, c[i]; i=8..15 -> lane 24+(i-8), c[i-8].
  float diag = 0.0f;
#pragma unroll
  for (int j = 0; j < 8; ++j) {
    if (lane == j)      diag = c[j];
    if (lane == 24 + j) diag = c[j];
  }
  float tot = wave_reduce_sum(diag);
  if (lane == 0) atomicAdd(sse, tot);
}

// ---------------------------------------------------------------------------
// Edge phase via the Tensor Data Mover in gather mode: one D# gathers 8 rows
// (32-bit row indices, 512B each) of the L2-resident diff[] into this wave's
// LDS slice. Two TDM ops per wave (src rows, dst rows), s_wait_tensorcnt,
// then rows are consumed with ds_load_b128 + wave32 max-reduce + f32 atomic.
// ---------------------------------------------------------------------------
__global__ void __launch_bounds__(TDM_WAVES * 32)
edge_kernel_tdm(const int* __restrict__ ei, const float* __restrict__ attr,
                const float* __restrict__ diff, float* __restrict__ node_w,
                int E, int N) {
  __shared__ float lbuf[TDM_WAVES * 2 * TDM_ROWS * D_FEAT];  // 32 KB

  const int lane = threadIdx.x & 31;
  const int wib  = __builtin_amdgcn_readfirstlane(threadIdx.x >> 5);
  const long long e0 =
      ((long long)blockIdx.x * TDM_WAVES + wib) * TDM_ROWS;

  float* bufA = lbuf + wib * (2 * TDM_ROWS * D_FEAT);
  float* bufB = bufA + TDM_ROWS * D_FEAT;

  // Wave-uniform row indices (contiguous in memory -> scalar loads).
  int si[TDM_ROWS], di[TDM_ROWS];
#pragma unroll
  for (int k = 0; k < TDM_ROWS; ++k) {
    long long e = e0 + k;
    int s = (e < E) ? ei[e] : 0;
    int d = (e < E) ? ei[(long long)E + e] : 0;
    si[k] = __builtin_amdgcn_readfirstlane(s);
    di[k] = __builtin_amdgcn_readfirstlane(d);
  }

  const unsigned long long gaddr = (unsigned long long)(uintptr_t)diff;
  const unsigned int ldsA = (unsigned int)(uintptr_t)bufA;  // low 32 bits = LDS offset
  const unsigned int ldsB = (unsigned int)(uintptr_t)bufB;
  const unsigned int g0w0 = 1u | (1u << 30) | (1u << 31);   // count=1, 32b idx, gather
  const unsigned int g0w3 = (unsigned int)(gaddr >> 32) | (2u << 30);  // addr[56:32], type=2

  u32x4 g0A = {g0w0, ldsA, (unsigned int)gaddr, g0w3};
  u32x4 g0B = {g0w0, ldsB, (unsigned int)gaddr, g0w3};
  // group1: data_size=4B; tensor_dim0=128, tensor_dim1=N, tile_dim0=128,
  //         tile_dim1=#indices=8, tensor_dim0_stride=128 (dim1 stride ignored).
  i32x8 g1 = {(int)(2u << 16),
              (int)((unsigned)(D_FEAT & 0xFFFF) << 16),
              (int)(((unsigned)N & 0xFFFF) << 16),
              (int)(((unsigned)N >> 16) | ((unsigned)D_FEAT << 16)),
              TDM_ROWS,
              D_FEAT,
              0, 0};
  i32x4 g2A = {si[0], si[1], si[2], si[3]};
  i32x4 g3A = {si[4], si[5], si[6], si[7]};
  i32x4 g2B = {di[0], di[1], di[2], di[3]};
  i32x4 g3B = {di[4], di[5], di[6], di[7]};

#if __clang_major__ >= 23
  i32x8 gz = {};
  __builtin_amdgcn_tensor_load_to_lds(g0A, g1, g2A, g3A, gz, 0);
  __builtin_amdgcn_tensor_load_to_lds(g0B, g1, g2B, g3B, gz, 0);
#else
  __builtin_amdgcn_tensor_load_to_lds(g0A, g1, g2A, g3A, 0);
  __builtin_amdgcn_tensor_load_to_lds(g0B, g1, g2B, g3B, 0);
#endif
  __builtin_amdgcn_s_wait_tensorcnt(0);

#pragma unroll
  for (int k = 0; k < TDM_ROWS; ++k) {
    long long e = e0 + k;
    if (e >= E) continue;  // wave-uniform
    float4 a = ((const float4*)(bufA + k * D_FEAT))[lane];
    float4 b = ((const float4*)(bufB + k * D_FEAT))[lane];
    float m = fmaxf(fmaxf(a.x - b.x, a.y - b.y), fmaxf(a.z - b.z, a.w - b.w));
    m = wave_reduce_max(m);
    if (lane == 0) atomicAdd(&node_w[di[k]], m / attr[e]);
  }
}

// Fallback (workspace too small for diff): direct VMEM gather of pred/data.
__global__ void edge_kernel_fused(const int* __restrict__ ei, const float* __restrict__ attr,
                                  const float* __restrict__ pred,
                                  const float* __restrict__ data,
                                  float* __restrict__ node_w, int E) {
  const int lane = threadIdx.x & 31;
  const long long wave =
      (long long)blockIdx.x * WAVES_PER_BLOCK + (threadIdx.x >> 5);
  const long long e0 = wave * EPW;
#pragma unroll
  for (int k = 0; k < EPW; ++k) {
    long long e = e0 + k;
    if (e >= E) continue;
    int s = ei[e], d = ei[(long long)E + e];
    float4 ps = ((const float4*)(pred + (size_t)s * D_FEAT))[lane];
    float4 qs = ((const float4*)(data + (size_t)s * D_FEAT))[lane];
    float4 pd = ((const float4*)(pred + (size_t)d * D_FEAT))[lane];
    float4 qd = ((const float4*)(data + (size_t)d * D_FEAT))[lane];
    float m = fmaxf(fmaxf((ps.x - qs.x) - (pd.x - qd.x), (ps.y - qs.y) - (pd.y - qd.y)),
                    fmaxf((ps.z - qs.z) - (pd.z - qd.z), (ps.w - qs.w) - (pd.w - qd.w)));
    m = wave_reduce_max(m);
    if (lane == 0) atomicAdd(&node_w[d], m / attr[e]);
  }
}

// ---------------------------------------------------------------------------
__global__ void clip_sum_kernel(const float* __restrict__ node_w,
                                float* __restrict__ nwsum, int N) {
  int i = blockIdx.x * blockDim.x + threadIdx.x;
  float v = (i < N) ? fminf(node_w[i], 1.0f) : 0.0f;
  v = wave_reduce_sum(v);
  if ((threadIdx.x & 31) == 0) atomicAdd(nwsum, v);
}

__global__ void finalize_kernel(const float* sse, const float* nwsum, float* out,
                                float inv_count) {
  out[0] = sse[0] * inv_count * nwsum[0];
}

// ---------------------------------------------------------------------------
extern "C" void kernel_launch(void* const* d_in, const int* in_sizes, int n_in,
                              void* d_out, int out_size, void* d_ws, size_t ws_size,
                              hipStream_t stream) {
  const float* pred = (const float*)d_in[0];
  const float* data = (const float*)d_in[1];
  const int* ei     = (const int*)d_in[2];
  const float* attr = (const float*)d_in[3];
  float* out        = (float*)d_out;

  const int ND = in_sizes[0];  // N * 128
  const int N  = ND / D_FEAT;
  const int E  = in_sizes[3];

  const size_t need_full = ((size_t)ND + (size_t)N + 2) * sizeof(float);
  const bool have_diff = ws_size >= need_full;

  float* diff   = (float*)d_ws;
  float* node_w = have_diff ? diff + (size_t)ND : (float*)d_ws;
  float* sse    = node_w + N;
  float* nwsum  = sse + 1;

  // 1) zero node_weight + accumulators (contiguous N+2 floats)
  {
    int n = N + 2;
    zero_ws_kernel<<<(n + 255) / 256, 256, 0, stream>>>(node_w, n);
  }

  // 2) diff + sum-of-squares via WMMA A*A^T diagonal trick
  {
    int tiles = ND / 64;  // even: = 2 tiles per node row
    if (have_diff)
      diff_sse_kernel<true><<<1250, 256, 0, stream>>>(pred, data, diff, sse, tiles);
    else
      diff_sse_kernel<false><<<1250, 256, 0, stream>>>(pred, data, diff, sse, tiles);
  }

  // 3) edge gather (TDM gather-mode rows -> LDS) + row-max + segment-sum
  if (have_diff) {
    int edgesPerBlock = TDM_WAVES * TDM_ROWS;
    int blocks = (E + edgesPerBlock - 1) / edgesPerBlock;
    edge_kernel_tdm<<<blocks, TDM_WAVES * 32, 0, stream>>>(ei, attr, diff, node_w, E, N);
  } else {
    int edgesPerBlock = WAVES_PER_BLOCK * EPW;
    int blocks = (E + edgesPerBlock - 1) / edgesPerBlock;
    edge_kernel_fused<<<blocks, 256, 0, stream>>>(ei, attr, pred, data, node_w, E);
  }

  // 4) clip + sum node weights
  clip_sum_kernel<<<(N + 255) / 256, 256, 0, stream>>>(node_w, nwsum, N);

  // 5) loss = mean(diff^2) * node_weight_sum
  finalize_kernel<<<1, 1, 0, stream>>>(sse, nwsum, out, 1.0f / (float)ND);
}